// LengthRegulator_29764123361505
// MI455X (gfx1250) — compile-verified
//
#include <hip/hip_runtime.h>
#include <hip/hip_bf16.h>

// LengthRegulator for MI455X (gfx1250, wave32).
// B=32, L=512, D=384, T_MAX=4096, EXPAND_SCALE=1.0 (round(1.0*int)=int), PAD=0.
// Memory-bound: ~201MB of output stores -> ~10us floor at 23.3 TB/s.
// Strategy:
//   K1: per-batch LDS scan of durations -> cum (workspace) + exp_dur tail of d_out.
//   K2: per block stage cum row + mask row (2KB each) into LDS via two
//       GLOBAL_LOAD_ASYNC_TO_LDS_B128 ops (single s_wait_asynccnt), 9-step
//       binary search in LDS per 16B chunk, b128 gather from L2-resident
//       sequence, non-temporal b128 store of the 201MB output stream.

typedef float v4f __attribute__((ext_vector_type(4)));
typedef int   v4i __attribute__((ext_vector_type(4)));

typedef __attribute__((address_space(1))) v4i* gv4i_p;  // global
typedef __attribute__((address_space(3))) v4i* lv4i_p;  // LDS

constexpr int Bc = 32;
constexpr int Lc = 512;
constexpr int Dc = 384;          // 96 float4 per row
constexpr int Tc = 4096;
constexpr int ROWS_PER_BLK = 16; // 16 rows * 96 chunks = 1536 chunks / 256 thr = 6 each

#if defined(__has_builtin)
#if __has_builtin(__builtin_amdgcn_global_load_async_to_lds_b128) && \
    __has_builtin(__builtin_amdgcn_s_wait_asynccnt)
#define HAVE_ASYNC_LDS 1
#endif
#endif
#ifndef HAVE_ASYNC_LDS
#define HAVE_ASYNC_LDS 0
#endif

// ---------------- Kernel 1: duration scan ----------------
// grid = B blocks, 512 threads. Hillis-Steele inclusive scan in LDS.
// Runs once over 16K ints -- negligible vs the 201MB expansion.
__global__ void lr_dur_scan(const int* __restrict__ dur,
                            int* __restrict__ cum,
                            float* __restrict__ expd_out) {
  __shared__ int s[Lc];
  const int b = blockIdx.x;
  const int t = threadIdx.x;
  const int e = dur[b * Lc + t];   // exp_dur == duration (scale 1.0, integer)
  s[t] = e;
  __syncthreads();
#pragma unroll
  for (int off = 1; off < Lc; off <<= 1) {
    int v = (t >= off) ? s[t - off] : 0;
    __syncthreads();
    s[t] += v;
    __syncthreads();
  }
  cum[b * Lc + t] = s[t];
  expd_out[b * Lc + t] = (float)e;
}

// ---------------- Kernel 2: expansion ----------------
// grid = B * (T/16) blocks, 256 threads (8 waves).
__global__ void lr_expand(const float* __restrict__ seq,
                          const int* __restrict__ mask,
                          const int* __restrict__ cum,
                          float* __restrict__ out) {
  __shared__ int s_cum[Lc];
  __shared__ int s_mask[Lc];
  const int tid = threadIdx.x;
  constexpr int tilesPerB = Tc / ROWS_PER_BLK;  // 256
  const int b  = blockIdx.x / tilesPerB;
  const int j0 = (blockIdx.x % tilesPerB) * ROWS_PER_BLK;

  const int* cum_row  = cum  + b * Lc;
  const int* mask_row = mask + b * Lc;

  // Stage 2x2KB rows into LDS: 128 lanes x b128 each, one async-cnt drain.
#if HAVE_ASYNC_LDS
  if (tid < 128) {
    __builtin_amdgcn_global_load_async_to_lds_b128(
        (gv4i_p)(cum_row + tid * 4),  (lv4i_p)(s_cum + tid * 4),  0, 0);
    __builtin_amdgcn_global_load_async_to_lds_b128(
        (gv4i_p)(mask_row + tid * 4), (lv4i_p)(s_mask + tid * 4), 0, 0);
  }
  __builtin_amdgcn_s_wait_asynccnt(0);
  __syncthreads();
#else
  if (tid < 128) {
    ((v4i*)s_cum)[tid]  = ((const v4i*)cum_row)[tid];
    ((v4i*)s_mask)[tid] = ((const v4i*)mask_row)[tid];
  }
  __syncthreads();
#endif

  const int total = s_cum[Lc - 1];
  const float* seq_b = seq + (size_t)b * Lc * Dc;
  float*       out_b = out + ((size_t)b * Tc + j0) * Dc;

#pragma unroll
  for (int c = 0; c < 6; ++c) {
    const int linear = c * 256 + tid;          // 0..1535
    const int r      = linear / 96;            // local output row (0..15)
    const int col4   = linear - r * 96;        // float4 column (0..95)
    const int j      = j0 + r;                 // global output slot

    // searchsorted(cum, j, side='right'): first index with cum[idx] > j.
    int lo = 0, hi = Lc;
#pragma unroll
    for (int step = 0; step < 9; ++step) {
      const int  mid = (lo + hi) >> 1;
      const bool le  = (s_cum[mid] <= j);
      lo = le ? mid + 1 : lo;
      hi = le ? hi : mid;
    }
    const int idx = (lo > Lc - 1) ? (Lc - 1) : lo;

    const bool live = (j < total) && (s_mask[idx] == 0);
    v4f val = {0.f, 0.f, 0.f, 0.f};
    if (live) {
      val = *(const v4f*)(seq_b + (size_t)idx * Dc + col4 * 4);
    }
    // Output is write-once streaming -> non-temporal, keep L2 for `seq` reuse.
    __builtin_nontemporal_store(val, (v4f*)(out_b + (size_t)r * Dc + col4 * 4));
  }
}

extern "C" void kernel_launch(void* const* d_in, const int* in_sizes, int n_in,
                              void* d_out, int out_size, void* d_ws, size_t ws_size,
                              hipStream_t stream) {
  const float* seq  = (const float*)d_in[0];
  const int*   dur  = (const int*)d_in[1];   // int64 durations delivered as int32 per harness convention
  const int*   mask = (const int*)d_in[2];   // bool mask as int32 (0/1)
  // d_in[3] = max_length scalar (== 4096, hardcoded as Tc)

  float* out  = (float*)d_out;
  float* expd = out + (size_t)Bc * Tc * Dc;  // second tuple output: exp_dur as f32
  int*   cum  = (int*)d_ws;                  // B*L int32 = 64 KB scratch

  lr_dur_scan<<<dim3(Bc), dim3(Lc), 0, stream>>>(dur, cum, expd);
  lr_expand<<<dim3(Bc * (Tc / ROWS_PER_BLK)), dim3(256), 0, stream>>>(seq, mask, cum, out);
}